// GroupLinear_68281390072017
// MI455X (gfx1250) — compile-verified
//
#include <hip/hip_runtime.h>

#define IN_F   1024
#define OUT_F  2048
#define NGROUP 8
#define NSAMP  8192

#define BM 64
#define BN 128
#define BK 32
#define LDK (BK + 2)            // padded LDS row stride (floats): conflict-free
#define MTILES 128              // NSAMP / BM worst case per group

typedef __attribute__((ext_vector_type(2))) float v2f;
typedef __attribute__((ext_vector_type(8))) float v8f;

// ---------------- bucketing kernels ----------------
// ws layout (ints): [0..7] counts, [8..15] offsets, [16..23] cursors, [24..24+NSAMP) perm

__global__ void gl_zero(int* ws) {
    if (threadIdx.x < 3 * NGROUP) ws[threadIdx.x] = 0;
}

__global__ void gl_hist(const int* __restrict__ idx, int* __restrict__ ws) {
    int n = blockIdx.x * blockDim.x + threadIdx.x;
    if (n < NSAMP) atomicAdd(&ws[idx[n] & (NGROUP - 1)], 1);
}

__global__ void gl_prefix(int* ws) {
    if (threadIdx.x == 0) {
        int s = 0;
        for (int g = 0; g < NGROUP; ++g) {
            ws[NGROUP + g]     = s;   // stable offsets
            ws[2 * NGROUP + g] = s;   // cursors for scatter
            s += ws[g];
        }
    }
}

__global__ void gl_scatter(const int* __restrict__ idx, int* __restrict__ ws) {
    int n = blockIdx.x * blockDim.x + threadIdx.x;
    if (n < NSAMP) {
        int g = idx[n] & (NGROUP - 1);
        int p = atomicAdd(&ws[2 * NGROUP + g], 1);
        ws[3 * NGROUP + p] = n;
    }
}

// ---------------- WMMA GEMM ----------------
// grid.x = NGROUP * MTILES, grid.y = OUT_F / BN, block = 256 (8 wave32)
__launch_bounds__(256)
__global__ void gl_gemm(const float* __restrict__ x, const float* __restrict__ W,
                        const float* __restrict__ bias, const int* __restrict__ ws,
                        float* __restrict__ out) {
    __shared__ float As[BM * LDK];
    __shared__ float Bs[BN * LDK];

    const int g     = blockIdx.x >> 7;        // / MTILES
    const int mt    = blockIdx.x & (MTILES - 1);
    const int count = ws[g];
    const int m0    = mt * BM;
    if (m0 >= count) return;                  // block-uniform: full waves exit

    const int  seg  = ws[NGROUP + g];
    const int* perm = ws + 3 * NGROUP;
    const int  n0   = blockIdx.y * BN;

    const int tid  = threadIdx.x;
    const int lane = tid & 31;
    const int wid  = tid >> 5;
    const int wm   = (wid & 1) * 32;          // wave M offset in block tile
    const int wn   = (wid >> 1) * 32;         // wave N offset in block tile

    const int hi   = lane >> 4;               // 0: lanes 0-15, 1: lanes 16-31
    const int mrow = lane & 15;
    const int koff = hi * 2;                  // A/B frag K base per half-wave

    const float* Wg = W + (size_t)(g * OUT_F + n0) * IN_F;

    v8f acc[2][2] = {};

    for (int kb = 0; kb < IN_F; kb += BK) {
        // ---- stage A tile (gathered rows, zero-padded) : 2 x float4 / thread
        #pragma unroll
        for (int i = 0; i < 2; ++i) {
            int fid = tid + 256 * i;          // [0,512)
            int m   = fid >> 3;
            int k4  = (fid & 7) << 2;
            float4 v = make_float4(0.f, 0.f, 0.f, 0.f);
            int local = m0 + m;
            if (local < count) {
                int s = perm[seg + local];
                v = *(const float4*)(x + (size_t)s * IN_F + kb + k4);
            }
            float* dst = &As[m * LDK + k4];
            dst[0] = v.x; dst[1] = v.y; dst[2] = v.z; dst[3] = v.w;
        }
        // ---- stage B tile (W rows for this group's output cols) : 4 x float4 / thread
        #pragma unroll
        for (int i = 0; i < 4; ++i) {
            int fid = tid + 256 * i;          // [0,1024)
            int n   = fid >> 3;
            int k4  = (fid & 7) << 2;
            float4 v = *(const float4*)(Wg + (size_t)n * IN_F + kb + k4);
            float* dst = &Bs[n * LDK + k4];
            dst[0] = v.x; dst[1] = v.y; dst[2] = v.z; dst[3] = v.w;
        }
        __syncthreads();

        // ---- 8 K-steps of V_WMMA_F32_16X16X4_F32, 2x2 tiles per wave
        #pragma unroll
        for (int k0 = 0; k0 < BK; k0 += 4) {
            v2f a[2], b[2];
            #pragma unroll
            for (int tm = 0; tm < 2; ++tm) {
                const float* p = &As[(wm + tm * 16 + mrow) * LDK + k0 + koff];
                v2f t = {p[0], p[1]};
                a[tm] = t;
            }
            #pragma unroll
            for (int tn = 0; tn < 2; ++tn) {
                const float* p = &Bs[(wn + tn * 16 + mrow) * LDK + k0 + koff];
                v2f t = {p[0], p[1]};
                b[tn] = t;
            }
            #pragma unroll
            for (int tm = 0; tm < 2; ++tm)
                #pragma unroll
                for (int tn = 0; tn < 2; ++tn)
                    acc[tm][tn] = __builtin_amdgcn_wmma_f32_16x16x4_f32(
                        false, a[tm], false, b[tn],
                        (short)0, acc[tm][tn], false, false);
        }
        __syncthreads();
    }

    // ---- store with fused bias, scattered back to original sample rows
    const float* bg = bias + g * OUT_F;
    #pragma unroll
    for (int tm = 0; tm < 2; ++tm) {
        #pragma unroll
        for (int r = 0; r < 8; ++r) {
            int local = m0 + wm + tm * 16 + r + hi * 8;
            if (local < count) {
                int s = perm[seg + local];
                #pragma unroll
                for (int tn = 0; tn < 2; ++tn) {
                    int col = n0 + wn + tn * 16 + mrow;
                    out[(size_t)s * OUT_F + col] = acc[tm][tn][r] + bg[col];
                }
            }
        }
    }
}

extern "C" void kernel_launch(void* const* d_in, const int* in_sizes, int n_in,
                              void* d_out, int out_size, void* d_ws, size_t ws_size,
                              hipStream_t stream) {
    const float* x    = (const float*)d_in[0];
    const int*   idx  = (const int*)d_in[1];
    const float* W    = (const float*)d_in[2];
    const float* bias = (const float*)d_in[3];
    float*       out  = (float*)d_out;
    int*         ws   = (int*)d_ws;

    gl_zero   <<<1, 32, 0, stream>>>(ws);
    gl_hist   <<<NSAMP / 256, 256, 0, stream>>>(idx, ws);
    gl_prefix <<<1, 1, 0, stream>>>(ws);
    gl_scatter<<<NSAMP / 256, 256, 0, stream>>>(idx, ws);

    dim3 grid(NGROUP * MTILES, OUT_F / BN);
    gl_gemm<<<grid, 256, 0, stream>>>(x, W, bias, ws, out);
}